// HyperNetResidualV2_63324997812748
// MI455X (gfx1250) — compile-verified
//
#include <hip/hip_runtime.h>
#include <hip/hip_bf16.h>

typedef __attribute__((ext_vector_type(2))) float v2f;
typedef __attribute__((ext_vector_type(8))) float v8f;

#define D_DIM  256
#define BN_EPS 1e-3f

// One wave computes 32 output columns (two 16-wide tiles) of y = x . W  (x: 1x256, W: 256x256 row-major).
// WMMA mapping: D(16x16) = A(16x4) x B(4x16) + C, with A row0 = x[d..d+3], rows 1..15 zero.
// A fp32 16x4 layout (ISA 7.12.2): V0 = K0 (lanes0-15, lane==M) / K2 (lanes16-31); V1 = K1 / K3.
//   -> row M=0 lives in lane 0 (K0,K1) and lane 16 (K2,K3); all other lanes zero.
// B fp32 4x16 layout (mirrors A, lane==N): V0 = K0 row (lanes0-15) / K2 row (lanes16-31); V1 = K1 / K3.
//   -> per lane: V0 = W[d + 2*(lane>>4)][e0 + (lane&15)], V1 = same + one row (256 floats).
// A-operand masking is done with a zero-padded LDS image (xz[256..511]==0): lanes that must carry
// zeros read from the zero region, so the ds_load itself produces the masked operand (no v_cndmask).
// Result: D[0][n] = acc[0] in lanes 0..15.
__device__ __forceinline__ void gemv_wmma(const float* __restrict__ Wb,   // W for this sample, 256x256
                                          const float* __restrict__ xz,   // LDS: [0..255]=x, [256..511]=0
                                          float*       __restrict__ ys,   // LDS: raw matvec output [256]
                                          int lane, int wave)
{
    const int  half  = lane >> 4;          // 0 -> K rows {0,1}, 1 -> K rows {2,3}
    const int  nIdx  = lane & 15;          // column within 16-wide tile
    const int  krow  = half * 2;
    const int  e0    = wave * 32;          // this wave owns columns [e0, e0+32)
    // lanes 0 and 16 carry A-matrix row M=0; all other lanes read from the zero pad
    const float* xa_base = xz + ((nIdx == 0) ? 0 : D_DIM) + krow;

    v8f acc0 = {0.f,0.f,0.f,0.f,0.f,0.f,0.f,0.f};
    v8f acc1 = acc0;

#pragma unroll 8
    for (int s = 0; s < D_DIM / 4; ++s) {
        const int d = s * 4;

        // A operand straight from LDS (zero pad handles masking); half-wave broadcast reads
        v2f A;
        A.x = xa_base[d];
        A.y = xa_base[d + 1];

        // B operands for the two 16-column tiles; 4 coalesced global_load_b32 per step
        const float* base = Wb + (size_t)(d + krow) * D_DIM + nIdx;
        v2f B0, B1;
        B0.x = base[e0];                 // row d+krow,   cols e0..e0+15 per half-wave
        B0.y = base[e0 + D_DIM];         // row d+krow+1
        B1.x = base[e0 + 16];            // second tile
        B1.y = base[e0 + 16 + D_DIM];

        acc0 = __builtin_amdgcn_wmma_f32_16x16x4_f32(false, A, false, B0, (short)0, acc0, false, false);
        acc1 = __builtin_amdgcn_wmma_f32_16x16x4_f32(false, A, false, B1, (short)0, acc1, false, false);
    }

    if (lane < 16) {                       // D row M=0 = acc[0], lanes 0..15
        ys[e0 + nIdx]      = acc0[0];
        ys[e0 + 16 + nIdx] = acc1[0];
    }
}

__global__ __launch_bounds__(D_DIM)
void HyperNetResidualV2_kernel(const float* __restrict__ xin,
                               const float* __restrict__ w1, const float* __restrict__ b1,
                               const float* __restrict__ w2, const float* __restrict__ b2,
                               const float* __restrict__ g1, const float* __restrict__ be1,
                               const float* __restrict__ m1, const float* __restrict__ v1,
                               const float* __restrict__ g2, const float* __restrict__ be2,
                               const float* __restrict__ m2, const float* __restrict__ v2,
                               float* __restrict__ out)
{
    __shared__ float xz[2 * D_DIM];        // [0..255] = activation, [256..511] = zeros (A-mask pad)
    __shared__ float ys[D_DIM];

    const int b    = blockIdx.x;           // one workgroup per sample: both blocks, no global sync
    const int tid  = threadIdx.x;          // feature index for elementwise phases
    const int lane = tid & 31;
    const int wave = tid >> 5;

    const float* W1b = w1 + (size_t)b * (D_DIM * D_DIM);
    const float* W2b = w2 + (size_t)b * (D_DIM * D_DIM);

    // ---- block 1 front-end: BN1 (inference) + ReLU -> LDS; init zero pad ----
    const float x0 = xin[(size_t)b * D_DIM + tid];
    const float s1 = g1[tid] * __builtin_amdgcn_rsqf(v1[tid] + BN_EPS);
    xz[tid]         = fmaxf((x0 - m1[tid]) * s1 + be1[tid], 0.0f);
    xz[D_DIM + tid] = 0.0f;

    // warm L2 for the phase-2 weight tile while we stream phase 1 (global_prefetch_b8)
    __builtin_prefetch(W2b + (size_t)tid * D_DIM, 0, 0);

    __syncthreads();

    // ---- block 1 GEMV: y1_raw = x . W1[b]  (streams 256 KB, WMMA fp32) ----
    gemv_wmma(W1b, xz, ys, lane, wave);
    __syncthreads();

    // ---- block 2 front-end: add b1, BN2 + ReLU -> LDS ----
    const float y1 = ys[tid] + b1[(size_t)b * D_DIM + tid];
    const float s2 = g2[tid] * __builtin_amdgcn_rsqf(v2[tid] + BN_EPS);
    const float h2 = fmaxf((y1 - m2[tid]) * s2 + be2[tid], 0.0f);
    __syncthreads();                       // all ys reads done before gemv overwrites it
    xz[tid] = h2;                          // zero pad untouched -> still valid
    __syncthreads();

    // ---- block 2 GEMV: y2_raw = h2 . W2[b] ----
    gemv_wmma(W2b, xz, ys, lane, wave);
    __syncthreads();

    // ---- residual: out = xin + y2_raw + b2 ----
    out[(size_t)b * D_DIM + tid] = x0 + ys[tid] + b2[(size_t)b * D_DIM + tid];
}

extern "C" void kernel_launch(void* const* d_in, const int* in_sizes, int n_in,
                              void* d_out, int out_size, void* d_ws, size_t ws_size,
                              hipStream_t stream) {
    // setup_inputs() order: xin, w1, b1, w2, b2, gamma1, beta1, mean1, var1, gamma2, beta2, mean2, var2
    const float* xin = (const float*)d_in[0];
    const float* w1  = (const float*)d_in[1];
    const float* b1  = (const float*)d_in[2];
    const float* w2  = (const float*)d_in[3];
    const float* b2  = (const float*)d_in[4];
    const float* g1  = (const float*)d_in[5];
    const float* be1 = (const float*)d_in[6];
    const float* m1  = (const float*)d_in[7];
    const float* v1  = (const float*)d_in[8];
    const float* g2  = (const float*)d_in[9];
    const float* be2 = (const float*)d_in[10];
    const float* m2  = (const float*)d_in[11];
    const float* v2  = (const float*)d_in[12];
    float* out = (float*)d_out;

    const int B = in_sizes[0] / D_DIM;     // 2048

    HyperNetResidualV2_kernel<<<B, D_DIM, 0, stream>>>(
        xin, w1, b1, w2, b2, g1, be1, m1, v1, g2, be2, m2, v2, out);
}